// NMSPostProcess_70463233458602
// MI455X (gfx1250) — compile-verified
//
#include <hip/hip_runtime.h>
#include <hip/hip_bf16.h>

#define NT        1024
#define NQ        900
#define NQPAD     912                   // 57 WMMA tiles x 16 rows (pad: unconditional D stores)
#define NCLS      91
#define NSC       (NQ * NCLS)           // 81900 scores per image
#define PRE_TOPK  10000
#define KEEP_K    100
#define IOU_THR   0.7f
#define MAXOWN    ((PRE_TOPK + NT - 1) / NT)   // 10 candidates per thread

typedef __attribute__((ext_vector_type(2))) float v2f;
typedef __attribute__((ext_vector_type(8))) float v8f;

// Order-preserving float -> uint key (monotonic, so top-k on keys == top-k on sigmoid(logit))
__device__ __forceinline__ unsigned fkey(float f) {
  unsigned u = __float_as_uint(f);
  unsigned m = (unsigned)(((int)u) >> 31) | 0x80000000u;
  return u ^ m;
}

__device__ __forceinline__ unsigned lds_off(const void* p) {
  // generic LDS pointer: addr[31:0] is the workgroup-relative LDS byte offset
  return (unsigned)(unsigned long long)p;
}

// wave32 (value,index) argmax combine; first-occurrence (min index) on ties
__device__ __forceinline__ void wreduce(float& v, int& i) {
  #pragma unroll
  for (int m = 16; m >= 1; m >>= 1) {
    float ov = __shfl_xor(v, m, 32);
    int   oi = __shfl_xor(i, m, 32);
    if (ov > v || (ov == v && oi < i)) { v = ov; i = oi; }
  }
}

__global__ __launch_bounds__(NT)
void nms_postprocess_fused(const float* __restrict__ logits,  // (bs,900,91)
                           const float* __restrict__ boxes,   // (bs,900,4) cxcywh
                           const float* __restrict__ tsz,     // (bs,2) h,w
                           float* __restrict__ out, int bs) {
  // ---- LDS state (~86 KB: one workgroup per WGP; MI455X has 320 KB/WGP) ----
  __shared__ __align__(16) float s_live[PRE_TOPK];  // live scores; first 3600 floats double as raw-box stage
  __shared__ float          s_box[NQPAD * 4];       // scaled xyxy per query box (padded to tile grid)
  __shared__ unsigned short s_bidx[PRE_TOPK];       // candidate -> query box index
  __shared__ unsigned char  s_lab[PRE_TOPK];        // candidate -> class label
  __shared__ unsigned       s_hist[256];
  __shared__ float          s_T[64];                // 4x16 WMMA B operand (transform, zero-padded)
  __shared__ float s_rv[32]; __shared__ int s_ri[32];
  __shared__ float s_pb[4];  __shared__ float s_pa;
  __shared__ unsigned s_sel, s_need, s_gc, s_ec;
  __shared__ float s_unit;

  const int tid  = threadIdx.x;
  const int lane = tid & 31, wave = tid >> 5;
  const int b    = blockIdx.x;
  const float* lg = logits + (size_t)b * NSC;
  const float* bx = boxes  + (size_t)b * NQ * 4;
  const float  H  = tsz[b * 2 + 0], W = tsz[b * 2 + 1];

  // ---- Phase 1a: async-stage raw cxcywh boxes into LDS (ASYNCcnt path), warm L2, build T ----
  float* s_raw = s_live;                       // alias: raw boxes live only until collection phase
  if (tid < NQ) {
    unsigned dst = lds_off(&s_raw[tid * 4]);
    unsigned long long src = (unsigned long long)(const void*)(bx + tid * 4);
    asm volatile("global_load_async_to_lds_b128 %0, %1, off"
                 :: "v"(dst), "v"(src) : "memory");
  }
  // Prefetch this image's logits into L2 (global_prefetch_b8); one lane per 128B line
  for (int i = tid * 32; i < NSC; i += NT * 32) __builtin_prefetch(lg + i, 0, 1);
  if (tid < 64) {                              // T[k][n]: xyxy = cxcywh * T, pre-scaled by (W,H)
    int k = tid >> 4, n = tid & 15;
    float v = 0.f;
    if (n < 4) {
      if      (k == 0) v = (n == 0 || n == 2) ? W : 0.f;          // cx
      else if (k == 1) v = (n == 1 || n == 3) ? H : 0.f;          // cy
      else if (k == 2) v = (n == 0) ? -0.5f * W : (n == 2 ?  0.5f * W : 0.f); // w
      else             v = (n == 1) ? -0.5f * H : (n == 3 ?  0.5f * H : 0.f); // h
    }
    s_T[k * 16 + n] = v;
  }
  asm volatile("s_wait_asynccnt 0" ::: "memory");
  __syncthreads();

  // ---- Phase 1b: (900x4)*(4x4) GEMM via v_wmma_f32_16x16x4_f32, 16-row tiles ----
  const int NTILE = NQPAD / 16;                // 57 tiles, wave-uniform loop (EXEC all ones)
  for (int tile = wave; tile < NTILE; tile += 32) {
    int m0   = tile * 16;
    int row  = m0 + (lane & 15);               // A 16x4: lanes 0-15 hold K0,K1; lanes 16-31 K2,K3
    int koff = (lane < 16) ? 0 : 2;
    v2f a;  a.x  = s_raw[row * 4 + koff];  a.y  = s_raw[row * 4 + koff + 1];
    int n16 = lane & 15;                       // B 4x16: V0 = K0/K2 halves, V1 = K1/K3 halves
    v2f bm; bm.x = s_T[((lane < 16) ? 0 : 2) * 16 + n16];
            bm.y = s_T[((lane < 16) ? 1 : 3) * 16 + n16];
    v8f c = {};
    v8f d = __builtin_amdgcn_wmma_f32_16x16x4_f32(false, a, false, bm, (short)0, c, false, false);
    if (n16 < 4) {                             // D 16x16: VGPR r -> rows r (lanes<16) / r+8 (lanes>=16)
      int mb = m0 + ((lane < 16) ? 0 : 8);
      #pragma unroll
      for (int r = 0; r < 8; ++r)              // unconditional: s_box padded to NQPAD rows
        s_box[(mb + r) * 4 + n16] = d[r];
    }
  }
  __syncthreads();

  // ---- Phase 2: exact top-10000 threshold via 4x8-bit radix histogram on logit keys ----
  if (tid == 0) s_need = PRE_TOPK;
  __syncthreads();
  unsigned prefix = 0;
  #pragma unroll 1
  for (int p = 0; p < 4; ++p) {
    for (int t = tid; t < 256; t += NT) s_hist[t] = 0;
    __syncthreads();
    const unsigned sh = 24 - 8 * p;
    const unsigned himask = (p == 0) ? 0u : (0xFFFFFFFFu << (sh + 8));
    for (int i = tid; i < NSC; i += NT) {
      unsigned k = fkey(lg[i]);
      if ((k & himask) == prefix) atomicAdd(&s_hist[(k >> sh) & 255u], 1u);
    }
    __syncthreads();
    if (tid == 0) {
      unsigned rem = s_need; int v = 255;
      for (; v > 0; --v) { unsigned c = s_hist[v]; if (c >= rem) break; rem -= c; }
      s_sel = (unsigned)v; s_need = rem;
    }
    __syncthreads();
    prefix |= s_sel << sh;
    __syncthreads();
  }
  const unsigned tkey = prefix;

  // ---- Phase 3: collect exactly 10000 candidates; sigmoid only for survivors ----
  if (tid == 0) { s_gc = 0; s_ec = 0; }
  __syncthreads();
  const unsigned need = s_need, G = PRE_TOPK - need;
  for (int i = tid; i < NSC; i += NT) {
    float x = lg[i];
    unsigned k = fkey(x);
    int slot = -1;
    if (k > tkey) slot = (int)atomicAdd(&s_gc, 1u);
    else if (k == tkey) {
      unsigned e = atomicAdd(&s_ec, 1u);
      if (e < need) slot = (int)(G + e);
    }
    if (slot >= 0) {
      int q = i / NCLS;
      s_bidx[slot] = (unsigned short)q;
      s_lab[slot]  = (unsigned char)(i - q * NCLS);
      s_live[slot] = 1.0f / (1.0f + __expf(-x));
    }
  }
  __syncthreads();

  // ---- Phase 4: per-class offset unit = max(gathered boxes) + 1 ----
  float mv = -1e30f;
  for (int slot = tid; slot < PRE_TOPK; slot += NT) {
    const float* B = &s_box[s_bidx[slot] * 4];
    mv = fmaxf(mv, fmaxf(fmaxf(B[0], B[1]), fmaxf(B[2], B[3])));
  }
  #pragma unroll
  for (int m = 16; m >= 1; m >>= 1) mv = fmaxf(mv, __shfl_xor(mv, m, 32));
  if (lane == 0) s_rv[wave] = mv;
  __syncthreads();
  if (wave == 0) {
    float u = s_rv[lane];
    #pragma unroll
    for (int m = 16; m >= 1; m >>= 1) u = fmaxf(u, __shfl_xor(u, m, 32));
    if (lane == 0) s_unit = u + 1.0f;
  }
  __syncthreads();
  const float unit = s_unit;

  // Pre-lift owned offset-boxes + areas into registers (<=10 per thread)
  float ox0[MAXOWN], oy0[MAXOWN], ox1[MAXOWN], oy1[MAXOWN], oar[MAXOWN];
  int n = 0;
  for (int slot = tid; slot < PRE_TOPK; slot += NT, ++n) {
    const float off = (float)s_lab[slot] * unit;
    const float* B = &s_box[s_bidx[slot] * 4];
    float x0 = B[0] + off, y0 = B[1] + off, x1 = B[2] + off, y1 = B[3] + off;
    ox0[n] = x0; oy0[n] = y0; ox1[n] = x1; oy1[n] = y1;
    oar[n] = (x1 - x0) * (y1 - y0);
  }

  // ---- Phase 5: 100 greedy NMS steps: workgroup argmax + IoU suppression, all in LDS ----
  const int lab_base = bs * KEEP_K, box_base = 2 * bs * KEEP_K;
  for (int it = 0; it < KEEP_K; ++it) {
    float bv = -__builtin_inff(); int bi = 0x7fffffff;
    n = 0;
    for (int slot = tid; slot < PRE_TOPK; slot += NT, ++n) {
      float v = s_live[slot];
      if (v > bv || (v == bv && slot < bi)) { bv = v; bi = slot; }
    }
    wreduce(bv, bi);
    if (lane == 0) { s_rv[wave] = bv; s_ri[wave] = bi; }
    __syncthreads();
    if (wave == 0) {                           // parallel cross-wave combine (shortens serial chain)
      float v = s_rv[lane]; int idx = s_ri[lane];
      wreduce(v, idx);
      if (lane == 0) {
        const int q = s_bidx[idx]; const int lab = s_lab[idx];
        const float* B = &s_box[q * 4];
        const int o = b * KEEP_K + it;
        out[o]            = v;            // score (picked while still live == original)
        out[lab_base + o] = (float)lab;   // label
        out[box_base + o * 4 + 0] = B[0];
        out[box_base + o * 4 + 1] = B[1];
        out[box_base + o * 4 + 2] = B[2];
        out[box_base + o * 4 + 3] = B[3];
        const float off = (float)lab * unit;
        s_pb[0] = B[0] + off; s_pb[1] = B[1] + off;
        s_pb[2] = B[2] + off; s_pb[3] = B[3] + off;
        s_pa    = (B[2] - B[0]) * (B[3] - B[1]);
      }
    }
    __syncthreads();
    const float p0 = s_pb[0], p1 = s_pb[1], p2 = s_pb[2], p3 = s_pb[3], pa = s_pa;
    n = 0;
    for (int slot = tid; slot < PRE_TOPK; slot += NT, ++n) {
      float lt0 = fmaxf(p0, ox0[n]), lt1 = fmaxf(p1, oy0[n]);
      float rb0 = fminf(p2, ox1[n]), rb1 = fminf(p3, oy1[n]);
      float w = fmaxf(rb0 - lt0, 0.f), h = fmaxf(rb1 - lt1, 0.f);
      float inter = w * h;
      float iou = inter / (pa + oar[n] - inter);
      if (iou > IOU_THR) s_live[slot] = -__builtin_inff();  // pivot self-suppresses (IoU=1), like reference
    }
    __syncthreads();
  }
}

extern "C" void kernel_launch(void* const* d_in, const int* in_sizes, int n_in,
                              void* d_out, int out_size, void* d_ws, size_t ws_size,
                              hipStream_t stream) {
  (void)n_in; (void)out_size; (void)d_ws; (void)ws_size;
  const float* logits = (const float*)d_in[0];
  const float* boxes  = (const float*)d_in[1];
  const float* tsz    = (const float*)d_in[2];
  float* out = (float*)d_out;
  const int bs = in_sizes[2] / 2;   // target_sizes is (bs, 2)
  nms_postprocess_fused<<<dim3(bs), dim3(NT), 0, stream>>>(logits, boxes, tsz, out, bs);
}